// SDEWiredLTCCell_8864812499232
// MI455X (gfx1250) — compile-verified
//
#include <hip/hip_runtime.h>

#define B_   16
#define K_   64
#define BK_  1024
#define IN_  128
#define H0_  192
#define H1_  64
#define UNF_ 6
#define EPS_ 1e-8f
#define DT_  (1.0f/6.0f)
#define SDT_ 0.40824829046386301637f
#define L2E_ 1.4426950408889634f

typedef float v2f __attribute__((ext_vector_type(2)));
typedef float v8f __attribute__((ext_vector_type(8)));

#if defined(__has_builtin)
#if __has_builtin(__builtin_amdgcn_global_load_async_to_lds_b128)
#define HAVE_ASYNC_LDS 1
#endif
#endif

// async-to-LDS builtin takes (global int4*, local int4*, imm offset, imm cpol)
typedef int v4i_ __attribute__((vector_size(16)));
typedef __attribute__((address_space(1))) v4i_ g_v4i;
typedef __attribute__((address_space(3))) v4i_ l_v4i;

__device__ __forceinline__ void wait_async0() {
#if defined(__has_builtin)
#if __has_builtin(__builtin_amdgcn_s_wait_asynccnt)
  __builtin_amdgcn_s_wait_asynccnt(0);
  return;
#else
  asm volatile("s_wait_asynccnt 0x0" ::: "memory");
  return;
#endif
#endif
}

// sigmoid(t) where u = -log2(e)*t :  1/(1 + 2^u)
__device__ __forceinline__ float sig_u(float u) {
  return __builtin_amdgcn_rcpf(1.0f + __builtin_amdgcn_exp2f(u));
}

// Pack (w, sig, mu, erev)[i] -> (-L2E*sig, L2E*sig*mu, relu(w), relu(w)*erev)
__global__ __launch_bounds__(256) void k_pack(const float* __restrict__ w,
                                              const float* __restrict__ sg,
                                              const float* __restrict__ mu,
                                              const float* __restrict__ er,
                                              float4* __restrict__ out, int n) {
  int i = blockIdx.x * blockDim.x + threadIdx.x;
  if (i < n) {
    float s = sg[i], m = mu[i];
    float wr = fmaxf(w[i], 0.0f);
    out[i] = make_float4(-L2E_ * s, L2E_ * s * m, wr, wr * er[i]);
  }
}

// Layer-0 input currents: only [B, H0] because x is broadcast over K particles.
__global__ __launch_bounds__(H0_) void k_input0(const float* __restrict__ x,
                                                const float4* __restrict__ ipk0,
                                                float* __restrict__ incur0) {
  __shared__ float xs[IN_];
  int h = threadIdx.x, b = blockIdx.x;
  if (h < IN_) xs[h] = x[b * IN_ + h];
  __syncthreads();
  float num = 0.f, den = 0.f;
  #pragma unroll 4
  for (int i = 0; i < IN_; ++i) {
    float4 q = ipk0[i * H0_ + h];
    float s  = sig_u(fmaf(q.x, xs[i], q.y));
    num = fmaf(q.w, s, num);
    den = fmaf(q.z, s, den);
  }
  incur0[b * H0_ + h]            = num;
  incur0[B_ * H0_ + b * H0_ + h] = den;
}

// Layer-0 unfold: 8 particles per block, all 6 steps locally; v in LDS.
#define P0 8
__global__ __launch_bounds__(H0_) void k_unfold0(
    const float* __restrict__ particles, const float* __restrict__ noise0,
    const float4* __restrict__ pk0, const float* __restrict__ incur0,
    const float* __restrict__ gleak, const float* __restrict__ vleak,
    const float* __restrict__ cm, const float* __restrict__ gdiff,
    float* __restrict__ s0, float* __restrict__ out_p) {
  __shared__ float vs[P0][H0_];
  int h = threadIdx.x;
  int bk0 = blockIdx.x * P0;
  int b = bk0 / K_;
  #pragma unroll
  for (int p = 0; p < P0; ++p)
    vs[p][h] = particles[(size_t)(bk0 + p) * (H0_ + H1_) + h];
  float gl   = fmaxf(gleak[h], 0.f);
  float glv  = gl * vleak[h];
  float invc = 1.0f / (fmaxf(cm[h], 0.f) + EPS_);
  float gd   = gdiff[h] * SDT_;
  float nin  = incur0[b * H0_ + h];
  float din  = incur0[B_ * H0_ + b * H0_ + h];
  __syncthreads();
  for (int s = 0; s < UNF_; ++s) {
    float accn[P0], accd[P0];
    #pragma unroll
    for (int p = 0; p < P0; ++p) { accn[p] = nin; accd[p] = din; }
    #pragma unroll 2
    for (int i = 0; i < H0_; ++i) {
      float4 q = pk0[i * H0_ + h];
      __builtin_prefetch(pk0 + (i + 8) * H0_ + h, 0, 0);  // global_prefetch_b8
      #pragma unroll
      for (int p = 0; p < P0; ++p) {
        float sg = sig_u(fmaf(q.x, vs[p][i], q.y));
        accn[p] = fmaf(q.w, sg, accn[p]);
        accd[p] = fmaf(q.z, sg, accd[p]);
      }
    }
    __syncthreads();
    #pragma unroll
    for (int p = 0; p < P0; ++p) {
      float v = vs[p][h];
      float d = (glv + accn[p] - (gl + accd[p]) * v) * invc;
      float nz = noise0[(size_t)s * BK_ * H0_ + (size_t)(bk0 + p) * H0_ + h];
      vs[p][h] = fmaf(d, DT_, fmaf(gd, nz, v));
    }
    __syncthreads();
  }
  #pragma unroll
  for (int p = 0; p < P0; ++p) {
    float v = vs[p][h];
    s0[(size_t)(bk0 + p) * H0_ + h] = v;
    out_p[(size_t)(bk0 + p) * (H0_ + H1_) + h] = v;
  }
}

// Layer-1 input currents: per-particle (s0 varies per particle), once (loop-invariant over unfolds).
__global__ __launch_bounds__(256) void k_input1(const float* __restrict__ s0,
                                                const float4* __restrict__ ipk1,
                                                float* __restrict__ incur1) {
  __shared__ float xs[4][H0_];
  int t = threadIdx.x;
  int p = t >> 6, h = t & 63;
  int bk0 = blockIdx.x * 4;
  for (int j = t; j < 4 * H0_; j += 256)
    xs[j / H0_][j % H0_] = s0[(size_t)bk0 * H0_ + j];
  __syncthreads();
  float num = 0.f, den = 0.f;
  #pragma unroll 4
  for (int i = 0; i < H0_; ++i) {
    float4 q = ipk1[i * H1_ + h];
    float s  = sig_u(fmaf(q.x, xs[p][i], q.y));
    num = fmaf(q.w, s, num);
    den = fmaf(q.z, s, den);
  }
  incur1[(size_t)(bk0 + p) * H1_ + h]             = num;
  incur1[BK_ * H1_ + (size_t)(bk0 + p) * H1_ + h] = den;
}

// Layer-1 unfold: 64-KB param pack staged once into dynamic LDS via
// GLOBAL_LOAD_ASYNC_TO_LDS_B128 (ASYNCcnt + s_wait_asynccnt), reused 6 steps x 8 particles.
#define P1 8
__global__ __launch_bounds__(H1_) void k_unfold1(
    const float* __restrict__ particles, const float* __restrict__ noise1,
    const float4* __restrict__ pk1, const float* __restrict__ incur1,
    const float* __restrict__ gleak, const float* __restrict__ vleak,
    const float* __restrict__ cm, const float* __restrict__ gdiff,
    float* __restrict__ s1, float* __restrict__ out_p) {
  extern __shared__ float4 pkl[];          // H1_*H1_ quads = 64 KB (dynamic LDS)
  __shared__ float vs[P1][H1_];
  int h = threadIdx.x;
  int bk0 = blockIdx.x * P1;
#if defined(HAVE_ASYNC_LDS)
  for (int j = h; j < H1_ * H1_; j += H1_)
    __builtin_amdgcn_global_load_async_to_lds_b128((g_v4i*)(pk1 + j),
                                                   (l_v4i*)(pkl + j), 0, 0);
#else
  for (int j = h; j < H1_ * H1_; j += H1_) pkl[j] = pk1[j];
#endif
  #pragma unroll
  for (int p = 0; p < P1; ++p)
    vs[p][h] = particles[(size_t)(bk0 + p) * (H0_ + H1_) + H0_ + h];
  float gl   = fmaxf(gleak[h], 0.f);
  float glv  = gl * vleak[h];
  float invc = 1.0f / (fmaxf(cm[h], 0.f) + EPS_);
  float gd   = gdiff[h] * SDT_;
#if defined(HAVE_ASYNC_LDS)
  wait_async0();
#endif
  __syncthreads();
  for (int s = 0; s < UNF_; ++s) {
    float accn[P1], accd[P1];
    #pragma unroll
    for (int p = 0; p < P1; ++p) {
      accn[p] = incur1[(size_t)(bk0 + p) * H1_ + h];
      accd[p] = incur1[BK_ * H1_ + (size_t)(bk0 + p) * H1_ + h];
    }
    #pragma unroll 2
    for (int i = 0; i < H1_; ++i) {
      float4 q = pkl[i * H1_ + h];         // ds_load_b128, conflict-free
      #pragma unroll
      for (int p = 0; p < P1; ++p) {
        float sg = sig_u(fmaf(q.x, vs[p][i], q.y));
        accn[p] = fmaf(q.w, sg, accn[p]);
        accd[p] = fmaf(q.z, sg, accd[p]);
      }
    }
    __syncthreads();
    #pragma unroll
    for (int p = 0; p < P1; ++p) {
      float v = vs[p][h];
      float d = (glv + accn[p] - (gl + accd[p]) * v) * invc;
      float nz = noise1[(size_t)s * BK_ * H1_ + (size_t)(bk0 + p) * H1_ + h];
      vs[p][h] = fmaf(d, DT_, fmaf(gd, nz, v));
    }
    __syncthreads();
  }
  #pragma unroll
  for (int p = 0; p < P1; ++p) {
    float v = vs[p][h];
    s1[(size_t)(bk0 + p) * H1_ + h] = v;
    out_p[(size_t)(bk0 + p) * (H0_ + H1_) + H0_ + h] = v;
  }
}

// softmax(log_weights) + weighted particle mean as block-diagonal GEMM
// [16 x 1024] . [1024 x 64] via V_WMMA_F32_16X16X4_F32.
// Block-diagonal A realized as unconditional LDS load * {0,1} mask (no EXEC games).
__global__ __launch_bounds__(128) void k_final(const float* __restrict__ lw,
                                               const float* __restrict__ s1,
                                               float* __restrict__ out_y,
                                               float* __restrict__ out_lw) {
  __shared__ float wts[B_ * K_];
  int t = threadIdx.x;
  if (t < B_) {
    int b = t;
    float mx = -3.0e38f;
    for (int k = 0; k < K_; ++k) mx = fmaxf(mx, lw[b * K_ + k]);
    float sum = 0.f;
    for (int k = 0; k < K_; ++k) sum += __builtin_amdgcn_exp2f(L2E_ * (lw[b * K_ + k] - mx));
    float inv = 1.0f / sum;
    for (int k = 0; k < K_; ++k)
      wts[b * K_ + k] = __builtin_amdgcn_exp2f(L2E_ * (lw[b * K_ + k] - mx)) * inv;
  }
  __syncthreads();
  int wave = t >> 5;          // n-tile 0..3 (16 cols each)
  int lane = t & 31;
  int m    = lane & 15;       // A row; C column
  int kh   = lane >> 4;       // k-half within 4-chunk
  int ncol = wave * 16 + (lane & 15);
  v8f c = {};
  for (int kc = 0; kc < BK_ / 4; ++kc) {
    int j0 = kc * 4 + kh * 2;
    // for this 4-wide chunk both A elements share one mask: rows 64m..64m+63 <=> kc>>4 == m
    float msk = ((kc >> 4) == m) ? 1.0f : 0.0f;
    v2f wv = *(const v2f*)&wts[j0];        // ds_load_b64, unconditional
    v2f a, bb;
    a.x = wv.x * msk;
    a.y = wv.y * msk;
    bb.x = s1[(size_t)j0 * H1_ + ncol];
    bb.y = s1[(size_t)(j0 + 1) * H1_ + ncol];
    c = __builtin_amdgcn_wmma_f32_16x16x4_f32(false, a, false, bb,
                                              (short)0, c, false, false);
  }
  #pragma unroll
  for (int r = 0; r < 8; ++r) {
    int mm = kh * 8 + r;
    out_y[mm * H1_ + ncol] = c[r];
  }
  __syncthreads();
  for (int j = t; j < B_ * K_; j += 128) out_lw[j] = lw[j];
}

extern "C" void kernel_launch(void* const* d_in, const int* in_sizes, int n_in,
                              void* d_out, int out_size, void* d_ws, size_t ws_size,
                              hipStream_t stream) {
  const float* x         = (const float*)d_in[0];
  const float* particles = (const float*)d_in[1];
  const float* lw        = (const float*)d_in[2];
  const float* noise0    = (const float*)d_in[3];
  const float* noise1    = (const float*)d_in[4];
  const float* gleak0 = (const float*)d_in[5],  *vleak0 = (const float*)d_in[6];
  const float* cm0    = (const float*)d_in[7];
  const float* iw0    = (const float*)d_in[8],  *isig0  = (const float*)d_in[9];
  const float* imu0   = (const float*)d_in[10], *ierev0 = (const float*)d_in[11];
  const float* w0     = (const float*)d_in[12], *sig0   = (const float*)d_in[13];
  const float* mu0    = (const float*)d_in[14], *erev0  = (const float*)d_in[15];
  const float* gdiff0 = (const float*)d_in[16];
  const float* gleak1 = (const float*)d_in[17], *vleak1 = (const float*)d_in[18];
  const float* cm1    = (const float*)d_in[19];
  const float* iw1    = (const float*)d_in[20], *isig1  = (const float*)d_in[21];
  const float* imu1   = (const float*)d_in[22], *ierev1 = (const float*)d_in[23];
  const float* w1     = (const float*)d_in[24], *sig1   = (const float*)d_in[25];
  const float* mu1    = (const float*)d_in[26], *erev1  = (const float*)d_in[27];
  const float* gdiff1 = (const float*)d_in[28];

  char* ws = (char*)d_ws;                       // ~2.85 MB total, 16B-aligned slices
  float4* pk0    = (float4*)(ws);               // 192*192 quads = 589824 B
  float4* pk1    = (float4*)(ws + 589824);      // 64*64   quads =  65536 B
  float4* ipk0   = (float4*)(ws + 655360);      // 128*192 quads = 393216 B
  float4* ipk1   = (float4*)(ws + 1048576);     // 192*64  quads = 196608 B
  float*  incur0 = (float*)(ws + 1245184);      // 2*16*192
  float*  incur1 = (float*)(ws + 1269760);      // 2*1024*64
  float*  s0     = (float*)(ws + 1794048);      // 1024*192
  float*  s1     = (float*)(ws + 2580480);      // 1024*64

  float* out_y  = (float*)d_out;                // [16,64]
  float* out_p  = out_y + B_ * H1_;             // [16,64,256]
  float* out_lw = out_p + (size_t)BK_ * (H0_ + H1_); // [16,64]

  k_pack<<<(H0_ * H0_ + 255) / 256, 256, 0, stream>>>(w0, sig0, mu0, erev0, pk0, H0_ * H0_);
  k_pack<<<(H1_ * H1_ + 255) / 256, 256, 0, stream>>>(w1, sig1, mu1, erev1, pk1, H1_ * H1_);
  k_pack<<<(IN_ * H0_ + 255) / 256, 256, 0, stream>>>(iw0, isig0, imu0, ierev0, ipk0, IN_ * H0_);
  k_pack<<<(H0_ * H1_ + 255) / 256, 256, 0, stream>>>(iw1, isig1, imu1, ierev1, ipk1, H0_ * H1_);

  k_input0<<<B_, H0_, 0, stream>>>(x, ipk0, incur0);
  k_unfold0<<<BK_ / P0, H0_, 0, stream>>>(particles, noise0, pk0, incur0,
                                          gleak0, vleak0, cm0, gdiff0, s0, out_p);
  k_input1<<<BK_ / 4, 256, 0, stream>>>(s0, ipk1, incur1);
  k_unfold1<<<BK_ / P1, H1_, 65536, stream>>>(particles, noise1, pk1, incur1,
                                              gleak1, vleak1, cm1, gdiff1, s1, out_p);
  k_final<<<1, 128, 0, stream>>>(lw, s1, out_y, out_lw);
}